// Yololoss_16604343567042
// MI455X (gfx1250) — compile-verified
//
#include <hip/hip_runtime.h>

// YOLO target-assignment IoU for gfx1250 (MI455X).
// 3 layers x 3 anchors x T targets independent work items; latency-bound.
// Targets staged in LDS via the CDNA5 async global->LDS path (ASYNCcnt).

#define NANCH 3
#define NLAYER 3
#define MAX_T 256          // reference uses T=200
#define RATIO_THRESH 4.0f

// ---- cooperative copy of the target table into LDS ------------------------
__device__ __forceinline__ void copy_targets_to_lds(const float* __restrict__ g,
                                                    float* lds, int tid,
                                                    int nthreads, int n) {
#if defined(__has_builtin) && __has_builtin(__builtin_amdgcn_global_load_async_to_lds_b32)
    // gfx1250 async copy: each lane copies 4 bytes from its global address to its LDS address, ASYNCcnt-tracked.
    for (int i = tid; i < n; i += nthreads) {
        __builtin_amdgcn_global_load_async_to_lds_b32(
            (__attribute__((address_space(1))) int*)(g + i),
            (__attribute__((address_space(3))) int*)(lds + i),
            /*offset=*/0, /*cpol=*/0);
    }
#if __has_builtin(__builtin_amdgcn_s_wait_asynccnt)
    __builtin_amdgcn_s_wait_asynccnt(0);
#else
    asm volatile("s_wait_asynccnt 0" ::: "memory");
#endif
#else
    for (int i = tid; i < n; i += nthreads) lds[i] = g[i];
#endif
}

__global__ __launch_bounds__(256) void yolo_iou_kernel(
    const float* __restrict__ pred0,   // (B,3,19,19,85)
    const float* __restrict__ pred1,   // (B,3,38,38,85)
    const float* __restrict__ pred2,   // (B,3,76,76,85)
    const float* __restrict__ targets, // (T,6) [b, cls, x, y, w, h]
    const float* __restrict__ anch0,   // (3,2)
    const float* __restrict__ anch1,
    const float* __restrict__ anch2,
    float* __restrict__ out,           // (3, 3*T)
    int T)
{
    __shared__ float sT[MAX_T * 6];
    __shared__ float sA[NLAYER * NANCH * 2];

    const int tid = threadIdx.x;

    // Stage targets (T*6 floats) and anchors (18 floats) in LDS.
    copy_targets_to_lds(targets, sT, tid, blockDim.x, T * 6);
    if (tid < NLAYER * NANCH * 2) {
        const int layer = tid / (NANCH * 2);
        const int off   = tid % (NANCH * 2);
        const float* ap = (layer == 0) ? anch0 : (layer == 1) ? anch1 : anch2;
        sA[tid] = ap[off];
    }
    __syncthreads();

    const int total = NLAYER * NANCH * T;
    const int g = blockIdx.x * blockDim.x + tid;
    if (g >= total) return;

    const int layer  = g / (NANCH * T);
    const int rem    = g - layer * (NANCH * T);
    const int anchor = rem / T;
    const int t      = rem - anchor * T;

    const int   G  = (layer == 0) ? 19 : (layer == 1) ? 38 : 76;   // gw == gh
    const float Gf = (float)G;
    const float* __restrict__ pp =
        (layer == 0) ? pred0 : (layer == 1) ? pred1 : pred2;

    // Target row from LDS.
    const int base = t * 6;
    const float bidx = sT[base + 0];
    const float x    = sT[base + 2];
    const float y    = sT[base + 3];
    const float w    = sT[base + 4];
    const float h    = sT[base + 5];

    const float aw = sA[(layer * NANCH + anchor) * 2 + 0];
    const float ah = sA[(layer * NANCH + anchor) * 2 + 1];

    // Grid-scaled target box.
    const float gx = x * Gf, gy = y * Gf;
    const float tw = w * Gf, th = h * Gf;

    // Integer cell (trunc toward zero like int32 cast), then clamp.
    int gi = (int)gx; gi = gi < 0 ? 0 : (gi > G - 1 ? G - 1 : gi);
    int gj = (int)gy; gj = gj < 0 ? 0 : (gj > G - 1 ? G - 1 : gj);

    const int b   = (int)bidx;
    const int idx = (((b * NANCH + anchor) * G + gj) * G + gi) * 85;

    // gfx1250 prefetch of the gathered pred cell (global_prefetch_b8).
    __builtin_prefetch(pp + idx, 0, 0);

    // Anchor-ratio keep mask.
    const float rw = tw / aw, rh = th / ah;
    const float rmax = fmaxf(fmaxf(rw, 1.0f / rw), fmaxf(rh, 1.0f / rh));
    const bool keep = rmax < RATIO_THRESH;

    // Gather the 4 box attributes of this prediction cell.
    const float ps0 = pp[idx + 0];
    const float ps1 = pp[idx + 1];
    const float ps2 = pp[idx + 2];
    const float ps3 = pp[idx + 3];

    // Predicted box: sigmoid(xy), exp(wh)*anchor.
    const float px = 1.0f / (1.0f + expf(-ps0));
    const float py = 1.0f / (1.0f + expf(-ps1));
    const float pw = expf(ps2) * aw;
    const float ph = expf(ps3) * ah;

    // Target box center = fractional offset within (clipped) cell.
    const float tbx = gx - (float)gi;
    const float tby = gy - (float)gj;

    // IoU (xywh, center-size).
    const float b1x1 = px - pw * 0.5f, b1x2 = px + pw * 0.5f;
    const float b1y1 = py - ph * 0.5f, b1y2 = py + ph * 0.5f;
    const float b2x1 = tbx - tw * 0.5f, b2x2 = tbx + tw * 0.5f;
    const float b2y1 = tby - th * 0.5f, b2y2 = tby + th * 0.5f;

    float iw = fminf(b1x2, b2x2) - fmaxf(b1x1, b2x1); iw = fmaxf(iw, 0.0f);
    float ih = fminf(b1y2, b2y2) - fmaxf(b1y1, b2y1); ih = fmaxf(ih, 0.0f);
    const float inter = iw * ih;
    const float uni   = pw * ph + tw * th - inter + 1e-9f;
    const float iou   = inter / uni;

    out[g] = keep ? iou : 0.0f;
}

extern "C" void kernel_launch(void* const* d_in, const int* in_sizes, int n_in,
                              void* d_out, int out_size, void* d_ws, size_t ws_size,
                              hipStream_t stream) {
    (void)n_in; (void)out_size; (void)d_ws; (void)ws_size;
    const float* pred0   = (const float*)d_in[0];
    const float* pred1   = (const float*)d_in[1];
    const float* pred2   = (const float*)d_in[2];
    const float* targets = (const float*)d_in[3];
    const float* anch0   = (const float*)d_in[4];
    const float* anch1   = (const float*)d_in[5];
    const float* anch2   = (const float*)d_in[6];
    float* out = (float*)d_out;

    const int T = in_sizes[3] / 6;           // 200 in the reference
    const int total = NLAYER * NANCH * T;    // 1800
    const int block = 256;                   // 8 wave32 waves
    const int grid  = (total + block - 1) / block;
    yolo_iou_kernel<<<grid, block, 0, stream>>>(pred0, pred1, pred2, targets,
                                                anch0, anch1, anch2, out, T);
}